// SimilarityBasedRelationEnhancer_35347580846912
// MI455X (gfx1250) — compile-verified
//
#include <hip/hip_runtime.h>
#include <math.h>
#include <stdint.h>

#define BB 64
#define RR 8192
#define DD 256
#define TILE_R 128
#define CHUNK_K 64
#define LDS_STRIDE 68   // 64 + 4 pad -> conflict-free ds_load_b64 A-fetch

typedef __attribute__((ext_vector_type(2))) float v2f;
typedef __attribute__((ext_vector_type(8))) float v8f;

__device__ __forceinline__ float sigf(float x) { return 1.0f / (1.0f + __expf(-x)); }
__device__ __forceinline__ int clampi(int v, int lo, int hi) {
    return v < lo ? lo : (v > hi ? hi : v);
}

// ---------------------------------------------------------------------------
// Kernel 0: gather query rows, compute L2 norm, store qrep and qn (normalized)
// ---------------------------------------------------------------------------
__global__ void __launch_bounds__(DD)
prep_query(const float* __restrict__ reps, const int* __restrict__ qrels,
           float* __restrict__ qrep_ws, float* __restrict__ qn_ws) {
    int b = blockIdx.x;
    int t = threadIdx.x;
    int q = clampi(qrels[b], 0, RR - 1);
    float v = reps[((size_t)b * RR + q) * DD + t];
    float ss = v * v;
    #pragma unroll
    for (int o = 16; o > 0; o >>= 1) ss += __shfl_down(ss, o, 32);
    __shared__ float part[8];
    __shared__ float tot;
    int lane = t & 31, w = t >> 5;
    if (lane == 0) part[w] = ss;
    __syncthreads();
    if (t == 0) {
        float s = 0.0f;
        #pragma unroll
        for (int i = 0; i < 8; ++i) s += part[i];
        tot = s;
    }
    __syncthreads();
    float inv = 1.0f / fmaxf(sqrtf(tot), 1e-12f);
    qrep_ws[b * DD + t] = v;
    qn_ws[b * DD + t]   = v * inv;
}

// ---------------------------------------------------------------------------
// Kernel 1: streaming pass over the 512MB tensor.
//   global --(global_load_async_to_lds_b128, ASYNCcnt)--> LDS tile
//   LDS    --(global_store_async_from_lds_b128)---------> out   (the copy)
//   LDS    --(ds_load_b64)--> V_WMMA_F32_16X16X4_F32     (sims mat-vec)
// Exactly one HBM read + one HBM write per element; no VGPR round-trip.
// ---------------------------------------------------------------------------
__global__ void __launch_bounds__(256)
copy_and_sims(const float* __restrict__ reps, const float* __restrict__ qn_ws,
              const int* __restrict__ qrels, float* __restrict__ out,
              float* __restrict__ sims_ws) {
    int b    = blockIdx.y;
    int row0 = blockIdx.x * TILE_R;
    int t    = threadIdx.x;
    int lane = t & 31;
    int wave = t >> 5;                 // 8 waves, 16 rows each

    __shared__ __align__(16) float smem[TILE_R * LDS_STRIDE];
    __shared__ float qn_s[DD];

    qn_s[t] = qn_ws[b * DD + t];

    const float* gin  = reps + ((size_t)b * RR + row0) * DD;
    float*       gout = out  + ((size_t)b * RR + row0) * DD;

    v8f acc;
    #pragma unroll
    for (int i = 0; i < 8; ++i) acc[i] = 0.0f;
    float nrm = 0.0f;

    for (int c = 0; c < DD / CHUNK_K; ++c) {
        int kc = c * CHUNK_K;
        __syncthreads();   // all waves done reading previous tile (ds + async stores issued)
        // WAR: previous chunk's async copy-out stores still read LDS; drain before
        // the new async loads overwrite it (async LDS accesses are unordered).
        asm volatile("s_wait_asynccnt 0x0" ::: "memory");
        // 128 rows x 64 cols = 2048 x b128 async transfers, 8 per thread
        #pragma unroll
        for (int i = 0; i < 8; ++i) {
            int idx = i * 256 + t;
            int row = idx >> 4;        // 16 float4 per row-chunk
            int c4  = idx & 15;
            uint64_t ga = (uint64_t)(uintptr_t)(gin + (size_t)row * DD + kc + (c4 << 2));
            uint32_t la = (uint32_t)(uintptr_t)(smem + row * LDS_STRIDE + (c4 << 2));
            asm volatile("global_load_async_to_lds_b128 %0, %1, off"
                         :: "v"(la), "v"(ga) : "memory");
        }
        asm volatile("s_wait_asynccnt 0x0" ::: "memory");   // tile landed in LDS
        __syncthreads();                                    // publish to all waves
        // fused copy: stream the tile back out LDS -> global (async engine)
        #pragma unroll
        for (int i = 0; i < 8; ++i) {
            int idx = i * 256 + t;
            int row = idx >> 4;
            int c4  = idx & 15;
            uint64_t ga = (uint64_t)(uintptr_t)(gout + (size_t)row * DD + kc + (c4 << 2));
            uint32_t la = (uint32_t)(uintptr_t)(smem + row * LDS_STRIDE + (c4 << 2));
            asm volatile("global_store_async_from_lds_b128 %0, %1, off"
                         :: "v"(ga), "v"(la) : "memory");
        }
        // WMMA compute: A-fetch lane<16 -> (row=lane, K+0..1); lane>=16 -> (row-16, K+2..3)
        const float2* a2 = reinterpret_cast<const float2*>(
            smem + (wave * 16 + (lane & 15)) * LDS_STRIDE + ((lane >> 4) << 1));
        #pragma unroll
        for (int k = 0; k < CHUNK_K; k += 4) {
            float2 a = a2[k >> 1];
            nrm = fmaf(a.x, a.x, nrm);
            nrm = fmaf(a.y, a.y, nrm);
            // B col 0 = qn chunk; lanes 0/16 carry K rows 0,1 (VGPR0) and 2,3 (VGPR1)
            float qa = qn_s[kc + k + (lane >> 4)];
            float qb = qn_s[kc + k + 2 + (lane >> 4)];
            bool n0 = (lane & 15) == 0;
            v2f av; av.x = a.x; av.y = a.y;
            v2f bv; bv.x = n0 ? qa : 0.0f; bv.y = n0 ? qb : 0.0f;
            acc = __builtin_amdgcn_wmma_f32_16x16x4_f32(
                false, av, false, bv, (short)0, acc, false, false);
        }
    }
    asm volatile("s_wait_asynccnt 0x0" ::: "memory");   // drain final copy-out
    // full row norms: halves live in lane L and L+16
    nrm += __shfl_xor(nrm, 16, 32);
    // D column N=0: lane 0 holds M=0..7 in acc[0..7]; lane 16 holds M=8..15
    int qi = clampi(qrels[b], 0, RR - 1);
    int mbase = (lane >> 4) << 3;      // 0 or 8
    #pragma unroll
    for (int m = 0; m < 8; ++m) {
        float nm = __shfl(nrm, mbase + m, 32);   // norm of row mbase+m
        float s  = acc[m] / fmaxf(sqrtf(nm), 1e-12f);
        int rglob = row0 + wave * 16 + mbase + m;
        if (rglob == qi) s = -1.0f;              // exclude self
        if ((lane & 15) == 0) sims_ws[(size_t)b * RR + rglob] = s;
    }
}

// ---------------------------------------------------------------------------
// Kernel 2: pointwise weights u_r and deterministic per-batch reductions.
// u_r = exp(s/T)*sigmoid(10(s-thr))*(1+scale*s) if s>thr else 0
// Z   = sum of exp(s/T) over masked rows (softmax denominator; cancels
//       everywhere except the 1e-8 epsilon term)
// ---------------------------------------------------------------------------
__global__ void __launch_bounds__(256)
compute_weights(const float* __restrict__ sims_ws, const float* __restrict__ thr_raw,
                const float* __restrict__ scale_p, const float* __restrict__ temp_p,
                float* __restrict__ u_ws, float* __restrict__ stats) {
    int b = blockIdx.x;
    int t = threadIdx.x;
    float thr   = sigf(thr_raw[0]);
    float temp  = fminf(fmaxf(temp_p[0], 0.1f), 10.0f);
    float scale = scale_p[0];
    float myU = 0.0f, myZ = 0.0f;
    int myV = 0;
    for (int r = t; r < RR; r += 256) {
        float s = sims_ws[(size_t)b * RR + r];
        bool  m = s > thr;
        float e = __expf(s / temp);
        float w = sigf((s - thr) * 10.0f);
        float u = m ? e * w * (1.0f + scale * s) : 0.0f;
        u_ws[(size_t)b * RR + r] = u;
        myU += u;
        myZ += m ? e : 0.0f;
        myV |= (int)m;
    }
    __shared__ float sU[256], sZ[256];
    __shared__ int   sV[256];
    sU[t] = myU; sZ[t] = myZ; sV[t] = myV;
    __syncthreads();
    for (int s = 128; s > 0; s >>= 1) {
        if (t < s) { sU[t] += sU[t + s]; sZ[t] += sZ[t + s]; sV[t] |= sV[t + s]; }
        __syncthreads();
    }
    if (t == 0) {
        stats[b * 4 + 0] = sU[0];
        stats[b * 4 + 1] = sZ[0];
        stats[b * 4 + 2] = (float)sV[0];
    }
}

// ---------------------------------------------------------------------------
// Kernel 3: weighted sum over rows with exact zero-weight pruning (u is
// block-uniform -> uniform branch skips the global row load entirely),
// blend with qrep, overwrite the query row in out.
// ---------------------------------------------------------------------------
__global__ void __launch_bounds__(256)
apply_enhance(const float* __restrict__ reps, const float* __restrict__ u_ws,
              const float* __restrict__ stats, const float* __restrict__ qrep_ws,
              const int* __restrict__ qrels, const float* __restrict__ str_raw,
              float* __restrict__ out) {
    int b = blockIdx.x;
    int t = threadIdx.x;                 // column of D
    float sumU = stats[b * 4 + 0];
    float Z    = stats[b * 4 + 1];
    bool  hasV = stats[b * 4 + 2] != 0.0f;
    int q = clampi(qrels[b], 0, RR - 1);
    float qv = qrep_ws[b * DD + t];
    float res;
    if (!hasV) {
        res = qv;
    } else {
        float acc = 0.0f;
        __shared__ float uS[256];
        for (int r0 = 0; r0 < RR; r0 += 256) {
            __syncthreads();
            uS[t] = u_ws[(size_t)b * RR + r0 + t];
            __syncthreads();
            for (int j = 0; j < 256; ++j) {
                float u = uS[j];
                if (u != 0.0f)   // uniform across block: prunes the row load
                    acc = fmaf(u, reps[((size_t)b * RR + r0 + j) * DD + t], acc);
            }
        }
        float strength = sigf(str_raw[0]) * 0.2f;
        float denom = sumU + 1e-8f * Z;   // == (sum adjusted) + eps, Z cancelled
        res = (1.0f - strength) * qv + strength * (acc / denom);
    }
    out[((size_t)b * RR + q) * DD + t] = res;
}

// ---------------------------------------------------------------------------
extern "C" void kernel_launch(void* const* d_in, const int* in_sizes, int n_in,
                              void* d_out, int out_size, void* d_ws, size_t ws_size,
                              hipStream_t stream) {
    const float* reps    = (const float*)d_in[0];
    const int*   qrels   = (const int*)d_in[1];
    const float* thr_raw = (const float*)d_in[2];
    const float* str_raw = (const float*)d_in[3];
    const float* scale_p = (const float*)d_in[4];
    const float* temp_p  = (const float*)d_in[5];
    float* out = (float*)d_out;

    float* ws      = (float*)d_ws;
    float* qrep_ws = ws;                        // B*D
    float* qn_ws   = qrep_ws + BB * DD;         // B*D
    float* sims_ws = qn_ws + BB * DD;           // B*R
    float* u_ws    = sims_ws + (size_t)BB * RR; // B*R
    float* stats   = u_ws + (size_t)BB * RR;    // B*4

    prep_query<<<BB, DD, 0, stream>>>(reps, qrels, qrep_ws, qn_ws);
    copy_and_sims<<<dim3(RR / TILE_R, BB), 256, 0, stream>>>(reps, qn_ws, qrels, out, sims_ws);
    compute_weights<<<BB, 256, 0, stream>>>(sims_ws, thr_raw, scale_p, temp_p, u_ws, stats);
    apply_enhance<<<BB, 256, 0, stream>>>(reps, u_ws, stats, qrep_ws, qrels, str_raw, out);
}